// E2E_DRO_Module_73461120631555
// MI455X (gfx1250) — compile-verified
//
#include <hip/hip_runtime.h>
#include <math.h>

typedef float v2f __attribute__((ext_vector_type(2)));
typedef float v8f __attribute__((ext_vector_type(8)));

#define TDIM 1024
#define NX 256
#define NY 64
#define GAMMA_C 0.05f
#define DYK_ITERS 200
#define FISTA_ITERS 400
#define POWER_ITERS 50
#define BIS_ITERS 40

// ---------------- wave32 / block reductions ----------------
__device__ __forceinline__ float waveSum(float v) {
#pragma unroll
  for (int o = 16; o > 0; o >>= 1) v += __shfl_xor(v, o, 32);
  return v;
}
__device__ __forceinline__ float waveMax(float v) {
#pragma unroll
  for (int o = 16; o > 0; o >>= 1) v = fmaxf(v, __shfl_xor(v, o, 32));
  return v;
}
__device__ __forceinline__ float waveMin(float v) {
#pragma unroll
  for (int o = 16; o > 0; o >>= 1) v = fminf(v, __shfl_xor(v, o, 32));
  return v;
}

template <int NW>
__device__ __forceinline__ float blockSum(float v, float* red) {
  v = waveSum(v);
  int lane = threadIdx.x & 31, wid = threadIdx.x >> 5;
  if (lane == 0) red[wid] = v;
  __syncthreads();
  float r = red[0];
#pragma unroll
  for (int i = 1; i < NW; ++i) r += red[i];
  __syncthreads();
  return r;
}
template <int NW>
__device__ __forceinline__ float blockMax(float v, float* red) {
  v = waveMax(v);
  int lane = threadIdx.x & 31, wid = threadIdx.x >> 5;
  if (lane == 0) red[wid] = v;
  __syncthreads();
  float r = red[0];
#pragma unroll
  for (int i = 1; i < NW; ++i) r = fmaxf(r, red[i]);
  __syncthreads();
  return r;
}
template <int NW>
__device__ __forceinline__ float blockMin(float v, float* red) {
  v = waveMin(v);
  int lane = threadIdx.x & 31, wid = threadIdx.x >> 5;
  if (lane == 0) red[wid] = v;
  __syncthreads();
  float r = red[0];
#pragma unroll
  for (int i = 1; i < NW; ++i) r = fminf(r, red[i]);
  __syncthreads();
  return r;
}

// Euclidean projection onto {x>=0, sum x = r}, vector distributed EPT per thread.
// Bisection on theta of f(theta)=sum(max(v-theta,0)) (monotone, piecewise linear):
// converges to the same theta as the sort-based reference within fp32 precision.
template <int NW, int EPT>
__device__ void projSimplex(const float* v, float r, float* out, float* red) {
  const float n = (float)(NW * 32 * EPT);
  float lmin = 3.4e38f, lmax = -3.4e38f;
#pragma unroll
  for (int j = 0; j < EPT; ++j) { lmin = fminf(lmin, v[j]); lmax = fmaxf(lmax, v[j]); }
  float lo = blockMin<NW>(lmin, red) - r / n;  // f(lo) >= r
  float hi = blockMax<NW>(lmax, red);          // f(hi) == 0 <= r
#pragma unroll 1
  for (int it = 0; it < BIS_ITERS; ++it) {
    float th = 0.5f * (lo + hi);
    float ls = 0.f;
#pragma unroll
    for (int j = 0; j < EPT; ++j) ls += fmaxf(v[j] - th, 0.f);
    float s = blockSum<NW>(ls, red);           // uniform across block
    if (s > r) lo = th; else hi = th;
  }
  float th = 0.5f * (lo + hi);
#pragma unroll
  for (int j = 0; j < EPT; ++j) out[j] = fmaxf(v[j] - th, 0.f);
}

// ---------------- f32 WMMA 16x16x4 fragments (CDNA5 ISA 7.12.2) ----------------
// A 16x4: lanes 0-15 -> {K0,K1}, lanes 16-31 -> {K2,K3}; row m = lane%16
__device__ __forceinline__ v2f loadA_frag(const float* A, int lda, int m0, int k0, int lane) {
  int half = lane >> 4, r = lane & 15;
  int k = k0 + 2 * half;
  const float* p = A + (size_t)(m0 + r) * lda + k;
  v2f a; a.x = p[0]; a.y = p[1];
  return a;
}
// B 4x16 row-major source: VGPR0 rows {k0,k0+2}, VGPR1 rows {k0+1,k0+3}; col n = n0+lane%16
__device__ __forceinline__ v2f loadB_frag(const float* B, int ldb, int k0, int n0, int lane) {
  int half = lane >> 4, r = lane & 15;
  int k = k0 + 2 * half;
  v2f b; b.x = B[(size_t)k * ldb + n0 + r]; b.y = B[(size_t)(k + 1) * ldb + n0 + r];
  return b;
}
// B given transposed in memory: B[k][n] = Bt[n][k]
__device__ __forceinline__ v2f loadBT_frag(const float* Bt, int ldbt, int k0, int n0, int lane) {
  int half = lane >> 4, r = lane & 15;
  int k = k0 + 2 * half;
  const float* p = Bt + (size_t)(n0 + r) * ldbt + k;
  v2f b; b.x = p[0]; b.y = p[1];
  return b;
}
__device__ __forceinline__ v8f wmma4f32(v2f a, v2f b, v8f c) {
  return __builtin_amdgcn_wmma_f32_16x16x4_f32(false, a, false, b, (short)0, c, false, false);
}

// ---------------- K1: Y_hat = X @ W^T + b ; ep = Y - Y_hat ----------------
__global__ __launch_bounds__(128) void k_pred(const float* __restrict__ X,
                                              const float* __restrict__ W,
                                              const float* __restrict__ bias,
                                              const float* __restrict__ Y,
                                              float* __restrict__ Yhat,
                                              float* __restrict__ ep) {
  int lane = threadIdx.x & 31, wid = threadIdx.x >> 5;
  int tile = blockIdx.x * 4 + wid;       // 256 tiles = 64 x 4
  int tm = tile >> 2, tn = tile & 3;
  int m0 = tm * 16, n0 = tn * 16;
  v8f acc = {};
  for (int k0 = 0; k0 < NX; k0 += 4) {
    v2f a = loadA_frag(X, NX, m0, k0, lane);
    v2f b = loadBT_frag(W, NX, k0, n0, lane);
    acc = wmma4f32(a, b, acc);
  }
  int half = lane >> 4, r = lane & 15;
  int n = n0 + r;
  float bn = bias[n];
#pragma unroll
  for (int v = 0; v < 8; ++v) {
    int m = m0 + v + 8 * half;
    float yh = acc[v] + bn;
    Yhat[(size_t)m * NY + n] = yh;
    ep[(size_t)m * NY + n] = Y[(size_t)m * NY + n] - yh;
  }
}

// ---------------- K2: PiT[t][o] = sum_j Z_star[t][j] * ep[o][j] ----------------
__global__ __launch_bounds__(128) void k_pit(const float* __restrict__ Zs,
                                             const float* __restrict__ ep,
                                             float* __restrict__ PiT) {
  int lane = threadIdx.x & 31, wid = threadIdx.x >> 5;
  int tile = blockIdx.x * 4 + wid;       // 4096 tiles = 64 x 64
  int tm = tile >> 6, tn = tile & 63;
  int m0 = tm * 16, n0 = tn * 16;
  v8f acc = {};
  for (int k0 = 0; k0 < NY; k0 += 4) {
    v2f a = loadA_frag(Zs, NY, m0, k0, lane);
    v2f b = loadBT_frag(ep, NY, k0, n0, lane);  // B[k][o] = ep[o][k]
    acc = wmma4f32(a, b, acc);
  }
  int half = lane >> 4, r = lane & 15;
#pragma unroll
  for (int v = 0; v < 8; ++v)
    PiT[(size_t)(m0 + v + 8 * half) * TDIM + n0 + r] = acc[v];
}

// ---------------- K3: Pprime[t][o] = P + gamma*(0.5*pi^2 - pi*s[t]) ----------------
__global__ __launch_bounds__(256) void k_pprime(const float* __restrict__ PiT,
                                                const float* __restrict__ P,
                                                float* __restrict__ Pp) {
  __shared__ float red[8];
  int t = blockIdx.x;
  const float* pi = PiT + (size_t)t * TDIM;
  const float* pr = P + (size_t)t * TDIM;
  float lpi[4], lp[4], acc = 0.f;
#pragma unroll
  for (int j = 0; j < 4; ++j) {
    int idx = threadIdx.x + j * 256;
    lpi[j] = pi[idx]; lp[j] = pr[idx];
    acc += lpi[j] * lp[j];
  }
  float s = blockSum<8>(acc, red);
#pragma unroll
  for (int j = 0; j < 4; ++j) {
    int idx = threadIdx.x + j * 256;
    Pp[(size_t)t * TDIM + idx] = lp[j] + GAMMA_C * (0.5f * lpi[j] * lpi[j] - lpi[j] * s);
  }
}

// ---------------- K4: Dykstra projection per row t ----------------
__global__ __launch_bounds__(256) void k_dykstra(const float* __restrict__ Pp,
                                                 const float* __restrict__ dptr,
                                                 float* __restrict__ Pnew) {
  __shared__ float red[8];
  int t = blockIdx.x;
  const float q = 1.0f / (float)TDIM;
  const float d = dptr[0];
  float x[4], iA[4], iB[4];
#pragma unroll
  for (int j = 0; j < 4; ++j) {
    x[j] = Pp[(size_t)t * TDIM + threadIdx.x + j * 256];
    iA[j] = 0.f; iB[j] = 0.f;
  }
  float ya[4], a[4], yb[4], aw[4], pj[4], w[4];
#pragma unroll 1
  for (int it = 0; it < DYK_ITERS; ++it) {
#pragma unroll
    for (int j = 0; j < 4; ++j) ya[j] = x[j] + iA[j];
    projSimplex<8, 4>(ya, 1.0f, a, red);
#pragma unroll
    for (int j = 0; j < 4; ++j) { iA[j] = ya[j] - a[j]; yb[j] = a[j] + iB[j]; }
    // proj_l1_ball(yb, q*1, d)
    float lsum = 0.f;
#pragma unroll
    for (int j = 0; j < 4; ++j) { w[j] = yb[j] - q; aw[j] = fabsf(w[j]); lsum += aw[j]; }
    float tot = blockSum<8>(lsum, red);
    projSimplex<8, 4>(aw, d, pj, red);
#pragma unroll
    for (int j = 0; j < 4; ++j) {
      float sg = (w[j] > 0.f) ? 1.f : ((w[j] < 0.f) ? -1.f : 0.f);
      float xb = (tot <= d) ? yb[j] : (q + sg * pj[j]);
      iB[j] = yb[j] - xb;
      x[j] = xb;
    }
  }
  projSimplex<8, 4>(x, 1.0f, a, red);
#pragma unroll
  for (int j = 0; j < 4; ++j)
    Pnew[(size_t)t * TDIM + threadIdx.x + j * 256] = a[j];
}

// ---------------- K5: Mu = P_new @ ep ----------------
__global__ __launch_bounds__(128) void k_mu(const float* __restrict__ Pnew,
                                            const float* __restrict__ ep,
                                            float* __restrict__ Mu) {
  int lane = threadIdx.x & 31, wid = threadIdx.x >> 5;
  int tile = blockIdx.x * 4 + wid;       // 256 tiles = 64 x 4
  int tm = tile >> 2, tn = tile & 3;
  int m0 = tm * 16, n0 = tn * 16;
  v8f acc = {};
  for (int k0 = 0; k0 < TDIM; k0 += 4) {
    v2f a = loadA_frag(Pnew, TDIM, m0, k0, lane);
    v2f b = loadB_frag(ep, NY, k0, n0, lane);
    acc = wmma4f32(a, b, acc);
  }
  int half = lane >> 4, r = lane & 15;
#pragma unroll
  for (int v = 0; v < 8; ++v)
    Mu[(size_t)(m0 + v + 8 * half) * NY + n0 + r] = acc[v];
}

// ---------------- K6: Q0[t] = S_t^T S_t, S[k][i] = sqrt(p[k])*(ep[k][i]-Mu[t][i]) ----------------
__global__ __launch_bounds__(128) void k_q0(const float* __restrict__ Pnew,
                                            const float* __restrict__ ep,
                                            const float* __restrict__ Mu,
                                            float* __restrict__ Q0) {
  __shared__ float sp[TDIM];
  int t = blockIdx.x;
  for (int i = threadIdx.x; i < TDIM; i += 128)
    sp[i] = sqrtf(Pnew[(size_t)t * TDIM + i]);
  __syncthreads();
  int lane = threadIdx.x & 31, wid = threadIdx.x >> 5;
  int half = lane >> 4, r = lane & 15;
  int I = wid;                    // wave handles tile-row I, tiles (I, J=0..3)
  int ci = I * 16 + r;
  float mu_i = Mu[(size_t)t * NY + ci];
  float mu_j[4];
#pragma unroll
  for (int J = 0; J < 4; ++J) mu_j[J] = Mu[(size_t)t * NY + J * 16 + r];
  v8f acc[4] = {};
  for (int k0 = 0; k0 < TDIM; k0 += 4) {
    int ka = k0 + 2 * half;
    float s0 = sp[ka], s1 = sp[ka + 1];
    v2f a;
    a.x = s0 * (ep[(size_t)ka * NY + ci] - mu_i);
    a.y = s1 * (ep[(size_t)(ka + 1) * NY + ci] - mu_i);
#pragma unroll
    for (int J = 0; J < 4; ++J) {
      int cj = J * 16 + r;
      v2f b;
      b.x = s0 * (ep[(size_t)ka * NY + cj] - mu_j[J]);
      b.y = s1 * (ep[(size_t)(ka + 1) * NY + cj] - mu_j[J]);
      acc[J] = wmma4f32(a, b, acc[J]);
    }
  }
#pragma unroll
  for (int J = 0; J < 4; ++J)
#pragma unroll
    for (int v = 0; v < 8; ++v)
      Q0[(size_t)t * (NY * NY) + (size_t)(I * 16 + v + 8 * half) * NY + J * 16 + r] = acc[J][v];
}

// ---------------- K7: power iteration + FISTA per t ----------------
__global__ __launch_bounds__(64) void k_solvez(const float* __restrict__ Q0,
                                               const float* __restrict__ Yhat,
                                               float* __restrict__ Z) {
  __shared__ float Q[NY * NY];
  __shared__ float xs[NY];
  __shared__ float red[2];
  int t = blockIdx.x;
  int i = threadIdx.x;
  for (int idx = i; idx < NY * NY; idx += 64)
    Q[idx] = Q0[(size_t)t * (NY * NY) + idx];
  float y = Yhat[(size_t)t * NY + i];
  __syncthreads();

  // power iteration: lambda_max estimate
  float v = 0.125f;  // 1/sqrt(64)
#pragma unroll 1
  for (int it = 0; it < POWER_ITERS; ++it) {
    xs[i] = v; __syncthreads();
    float w = 0.f;
#pragma unroll
    for (int j = 0; j < NY; ++j) w += Q[i * NY + j] * xs[j];
    __syncthreads();
    float nrm = sqrtf(blockSum<2>(w * w, red));
    v = w / (nrm + 1e-12f);
  }
  xs[i] = v; __syncthreads();
  float w = 0.f;
#pragma unroll
  for (int j = 0; j < NY; ++j) w += Q[i * NY + j] * xs[j];
  __syncthreads();
  float lam = blockSum<2>(v * w, red);
  float L = 20.0f * lam * 1.05f + 1e-6f;

  // FISTA with simplex projection
  float z = 1.0f / (float)NY, x = z, tt = 1.0f;
#pragma unroll 1
  for (int it = 0; it < FISTA_ITERS; ++it) {
    xs[i] = x; __syncthreads();
    float g = 0.f;
#pragma unroll
    for (int j = 0; j < NY; ++j) g += Q[i * NY + j] * xs[j];
    __syncthreads();
    g = 20.0f * g - y;
    float u = x - g / L;
    float zn;
    projSimplex<2, 1>(&u, 1.0f, &zn, red);
    float tn = 0.5f * (1.0f + sqrtf(1.0f + 4.0f * tt * tt));
    x = zn + ((tt - 1.0f) / tn) * (zn - z);
    z = zn; tt = tn;
  }
  Z[(size_t)t * NY + i] = z;
}

// ---------------- launch ----------------
extern "C" void kernel_launch(void* const* d_in, const int* in_sizes, int n_in,
                              void* d_out, int out_size, void* d_ws, size_t ws_size,
                              hipStream_t stream) {
  const float* X  = (const float*)d_in[0];   // (T, NX)
  const float* Y  = (const float*)d_in[1];   // (T, NY)
  const float* P  = (const float*)d_in[2];   // (T, T)
  const float* Zs = (const float*)d_in[3];   // (T, NY)
  const float* W  = (const float*)d_in[4];   // (NY, NX)
  const float* b  = (const float*)d_in[5];   // (NY)
  const float* d  = (const float*)d_in[6];   // (1)

  float* Z_out  = (float*)d_out;                         // (T, NY)
  float* P_out  = Z_out + (size_t)TDIM * NY;             // (T, T)
  float* Yh_out = P_out + (size_t)TDIM * TDIM;           // (T, NY)

  float* ws  = (float*)d_ws;
  float* ep  = ws;                                   // T*NY      = 65536
  float* PiT = ep + (size_t)TDIM * NY;               // T*T       = 1048576
  float* Pp  = PiT + (size_t)TDIM * TDIM;            // T*T       = 1048576
  float* Mu  = Pp + (size_t)TDIM * TDIM;             // T*NY      = 65536
  float* Q0  = Mu + (size_t)TDIM * NY;               // T*NY*NY   = 4194304

  k_pred   <<<64,   128, 0, stream>>>(X, W, b, Y, Yh_out, ep);
  k_pit    <<<1024, 128, 0, stream>>>(Zs, ep, PiT);
  k_pprime <<<1024, 256, 0, stream>>>(PiT, P, Pp);
  k_dykstra<<<1024, 256, 0, stream>>>(Pp, d, P_out);
  k_mu     <<<64,   128, 0, stream>>>(P_out, ep, Mu);
  k_q0     <<<1024, 128, 0, stream>>>(P_out, ep, Mu, Q0);
  k_solvez <<<1024,  64, 0, stream>>>(Q0, Yh_out, Z_out);
}